// Retriever_65816078844452
// MI455X (gfx1250) — compile-verified
//
#include <hip/hip_runtime.h>
#include <math.h>

typedef __attribute__((ext_vector_type(2))) float v2f;
typedef __attribute__((ext_vector_type(4))) float v4f;
typedef __attribute__((ext_vector_type(8))) float v8f;

#define NQ     4096
#define DIM    1024
#define NK     100000
#define TOPK   32
#define QPITCH 1028   // rows 16B-aligned; A-frag b64 reads hit banks (4m+2h+e) = 0..63, conflict-free

__device__ __forceinline__ unsigned ordkey(float f) {
    unsigned u = __float_as_uint(f);
    return (u & 0x80000000u) ? ~u : (u | 0x80000000u);
}

__global__ __launch_bounds__(256)
void retriever_topk_wmma(const float* __restrict__ query,
                         const float* __restrict__ knw,
                         int* __restrict__ out)
{
    __shared__ float qtile[16 * QPITCH];        // 65792 B : Q tile, row pitch 1028
    __shared__ float simb [8 * 16 * 17];        //  8704 B : per-wave 16x16 sim scratch
    __shared__ float lval [8 * TOPK * 32];      // 32768 B : [wave][entry][lane] values
    __shared__ int   lidx [8 * TOPK * 32];      // 32768 B : [wave][entry][lane] indices

    const int tid   = threadIdx.x;
    const int w     = tid >> 5;        // wave id 0..7
    const int lane  = tid & 31;
    const int m     = lane & 15;       // matrix row (A) / tile column (B) / scan row
    const int h     = lane >> 4;       // K-half for A/B frags, column-half for scan
    const int qbase = blockIdx.x * 16;

    // ---- stage Q tile into LDS (coalesced float4, rows 16B-aligned) ----
    for (int i = tid; i < 16 * (DIM / 4); i += 256) {
        int row = i >> 8;              // 256 float4 per row
        int c4  = i & 255;
        v4f v = *(const v4f*)(query + (size_t)(qbase + row) * DIM + 4 * c4);
        *(v4f*)(qtile + row * QPITCH + 4 * c4) = v;
    }

    // ---- init per-lane top-k lists ----
#pragma unroll
    for (int e = 0; e < TOPK; ++e) {
        lval[(w * TOPK + e) * 32 + lane] = -INFINITY;
        lidx[(w * TOPK + e) * 32 + lane] = 0;
    }
    float minv = -INFINITY;

    __syncthreads();

    const float* aptr = qtile + m * QPITCH + 2 * h;   // A fragment source (LDS)
    float* sb = simb + w * (16 * 17);

    const int NITER = (NK + 511) / 512;               // 512 knowledge cols / iter / WG
    for (int it = 0; it < NITER; ++it) {
        const int nbase = it * 512 + w * 64;

        // clamped B base pointers for the 4 column tiles of this wave
        const float* bptr0; const float* bptr1; const float* bptr2; const float* bptr3;
        {
            int n0 = nbase +  0 + m; if (n0 > NK - 1) n0 = NK - 1;
            int n1 = nbase + 16 + m; if (n1 > NK - 1) n1 = NK - 1;
            int n2 = nbase + 32 + m; if (n2 > NK - 1) n2 = NK - 1;
            int n3 = nbase + 48 + m; if (n3 > NK - 1) n3 = NK - 1;
            bptr0 = knw + (size_t)n0 * DIM + 2 * h;
            bptr1 = knw + (size_t)n1 * DIM + 2 * h;
            bptr2 = knw + (size_t)n2 * DIM + 2 * h;
            bptr3 = knw + (size_t)n3 * DIM + 2 * h;
        }

        v8f acc0 = {}, acc1 = {}, acc2 = {}, acc3 = {};

        // ---- GEMM: K loop, 4 WMMAs per A fragment (A reuse x4), deep load pipeline ----
#pragma unroll 8
        for (int kk = 0; kk < DIM / 4; ++kk) {
            v2f a  = *(const v2f*)(aptr  + kk * 4);
            v2f b0 = *(const v2f*)(bptr0 + kk * 4);
            v2f b1 = *(const v2f*)(bptr1 + kk * 4);
            v2f b2 = *(const v2f*)(bptr2 + kk * 4);
            v2f b3 = *(const v2f*)(bptr3 + kk * 4);
            acc0 = __builtin_amdgcn_wmma_f32_16x16x4_f32(false, a, false, b0, (short)0, acc0, false, false);
            acc1 = __builtin_amdgcn_wmma_f32_16x16x4_f32(false, a, false, b1, (short)0, acc1, false, false);
            acc2 = __builtin_amdgcn_wmma_f32_16x16x4_f32(false, a, false, b2, (short)0, acc2, false, false);
            acc3 = __builtin_amdgcn_wmma_f32_16x16x4_f32(false, a, false, b3, (short)0, acc3, false, false);
        }

        // ---- per-tile: spill C to LDS, threshold-gated top-k insert ----
#pragma unroll
        for (int t = 0; t < 4; ++t) {
            v8f c = (t == 0) ? acc0 : (t == 1) ? acc1 : (t == 2) ? acc2 : acc3;
            // C layout: VGPR r -> row (r + 8*h), lane%16 -> col
#pragma unroll
            for (int r = 0; r < 8; ++r)
                sb[(r + 8 * h) * 17 + m] = c[r];
            __builtin_amdgcn_wave_barrier();

            const int ncol0 = nbase + t * 16 + 8 * h;
#pragma unroll
            for (int j = 0; j < 8; ++j) {
                float v = sb[m * 17 + 8 * h + j];
                int   n = ncol0 + j;
                if (n < NK && v > minv) {
                    // replace current min entry, then re-derive the min
                    int am = 0;
                    float mv = lval[(w * TOPK + 0) * 32 + lane];
                    for (int e = 1; e < TOPK; ++e) {
                        float x = lval[(w * TOPK + e) * 32 + lane];
                        if (x < mv) { mv = x; am = e; }
                    }
                    lval[(w * TOPK + am) * 32 + lane] = v;
                    lidx[(w * TOPK + am) * 32 + lane] = n;
                    float nm = v;
                    for (int e = 0; e < TOPK; ++e)
                        nm = fminf(nm, lval[(w * TOPK + e) * 32 + lane]);
                    minv = nm;
                }
            }
            __builtin_amdgcn_wave_barrier();   // before next tile reuses sb
        }
    }

    __syncthreads();

    // ---- merge 512 candidates/row -> final top-32 (wave w handles rows w, w+8) ----
#pragma unroll
    for (int rr = 0; rr < 2; ++rr) {
        const int row = w + 8 * rr;
        unsigned long long cand[16];
#pragma unroll
        for (int i = 0; i < 16; ++i) {
            int c  = lane * 16 + i;            // 0..511
            int wp = c >> 6;
            int hh = (c >> 5) & 1;
            int e  = c & 31;
            int sl = hh * 16 + row;            // owning lane within wave wp
            float    v = lval[(wp * TOPK + e) * 32 + sl];
            unsigned n = (unsigned)lidx[(wp * TOPK + e) * 32 + sl];
            // key: bigger sim first, ties -> smaller knowledge index
            cand[i] = ((unsigned long long)ordkey(v) << 32) | (unsigned)(~n);
        }
        for (int r = 0; r < TOPK; ++r) {
            unsigned long long best = cand[0];
#pragma unroll
            for (int i = 1; i < 16; ++i) best = (cand[i] > best) ? cand[i] : best;
#pragma unroll
            for (int off = 16; off > 0; off >>= 1) {
                unsigned long long o = __shfl_xor(best, off, 32);
                if (o > best) best = o;
            }
            if (lane == 0)
                out[(qbase + row) * TOPK + r] = (int)(~(unsigned)best);
#pragma unroll
            for (int i = 0; i < 16; ++i)
                if (cand[i] == best) cand[i] = 0ull;   // retire winner
        }
    }
}

extern "C" void kernel_launch(void* const* d_in, const int* in_sizes, int n_in,
                              void* d_out, int out_size, void* d_ws, size_t ws_size,
                              hipStream_t stream) {
    (void)in_sizes; (void)n_in; (void)d_ws; (void)ws_size; (void)out_size;
    const float* q   = (const float*)d_in[0];
    const float* knw = (const float*)d_in[1];
    int*         out = (int*)d_out;
    dim3 grid(NQ / 16);   // 256 workgroups, 16 query rows each
    dim3 block(256);      // 8 wave32 waves
    hipLaunchKernelGGL(retriever_topk_wmma, grid, block, 0, stream, q, knw, out);
}